// Block_19301583028789
// MI455X (gfx1250) — compile-verified
//
#include <hip/hip_runtime.h>
#include <hip/hip_bf16.h>
#include <math.h>

// Problem constants (match reference)
#define Bdim 16
#define Hdim 64
#define Ldim 16384
#define Ndim 64
#define CDdim 32

// Parallel-scan partitioning of the L dimension
#define Pdim 16
#define SEG  (Ldim / Pdim)   // 1024

typedef __attribute__((ext_vector_type(2))) float v2f;
typedef __attribute__((ext_vector_type(8))) float v8f;

__device__ __forceinline__ float gelu_exact(float v) {
    // 0.5 * v * (1 + erf(v / sqrt(2)))
    return 0.5f * v * (1.0f + erff(v * 0.70710678118654752f));
}

// ---------------------------------------------------------------------------
// Kernel 1: S4D parameter precompute.
// lam  = exp(dt*A)                      (per-step decay)
// lamS = exp(SEG*dt*A)                  (per-segment decay, closed form)
// c2   = 2 * C * (exp(dt*A) - 1) / A    (output coefficient)
// ---------------------------------------------------------------------------
__global__ void s4d_params_kernel(const float* __restrict__ log_dt,
                                  const float* __restrict__ log_A_real,
                                  const float* __restrict__ A_imag,
                                  const float* __restrict__ C_re,
                                  const float* __restrict__ C_im,
                                  float* __restrict__ lam_re,
                                  float* __restrict__ lam_im,
                                  float* __restrict__ lamS_re,
                                  float* __restrict__ lamS_im,
                                  float* __restrict__ c2_re,
                                  float* __restrict__ c2_im) {
    int idx = blockIdx.x * blockDim.x + threadIdx.x;
    if (idx >= Hdim * Ndim) return;
    int h = idx / Ndim;
    float dt = expf(log_dt[h]);
    float Ar = -expf(log_A_real[idx]);
    float Ai = A_imag[idx];
    // lam = exp(dt*A)
    float er = expf(Ar * dt);
    float wr = er * cosf(Ai * dt);
    float wi = er * sinf(Ai * dt);
    // lamS = exp(SEG*dt*A)
    float ts = dt * (float)SEG;
    float erS = expf(Ar * ts);
    lamS_re[idx] = erS * cosf(Ai * ts);
    lamS_im[idx] = erS * sinf(Ai * ts);
    // (lam - 1) / A  via conj(A)/|A|^2
    float nr = wr - 1.0f, ni = wi;
    float inv = 1.0f / (Ar * Ar + Ai * Ai);
    float qr = (nr * Ar + ni * Ai) * inv;
    float qi = (ni * Ar - nr * Ai) * inv;
    float cr = C_re[idx], ci = C_im[idx];
    lam_re[idx] = wr;
    lam_im[idx] = wi;
    c2_re[idx] = 2.0f * (cr * qr - ci * qi);
    c2_im[idx] = 2.0f * (cr * qi + ci * qr);
}

// ---------------------------------------------------------------------------
// Kernel 2: FiLM: gb[b,j] = cond[b,:] . film_W[j,:] + film_b[j]
// ---------------------------------------------------------------------------
__global__ void film_kernel(const float* __restrict__ cond,
                            const float* __restrict__ film_W,
                            const float* __restrict__ film_b,
                            float* __restrict__ g_out,   // [B,H]
                            float* __restrict__ b_out) { // [B,H]
    int idx = blockIdx.x * blockDim.x + threadIdx.x;
    if (idx >= Bdim * 2 * Hdim) return;
    int b = idx / (2 * Hdim);
    int j = idx % (2 * Hdim);
    float acc = film_b[j];
    const float* cb = cond + b * CDdim;
    const float* wj = film_W + (size_t)j * CDdim;
#pragma unroll
    for (int c = 0; c < CDdim; ++c) acc = fmaf(cb[c], wj[c], acc);
    if (j < Hdim) g_out[b * Hdim + j] = acc;
    else          b_out[b * Hdim + (j - Hdim)] = acc;
}

// ---------------------------------------------------------------------------
// Kernel 3: channel-mix GEMM via V_WMMA_F32_16X16X4_F32, fused bias + GELU.
//   u[b,g,l] = gelu( sum_h W[g,h] * x[b,h,l] + b_lin[g] )
// One wave computes a full 64(g) x 16(l) column block: 4 accumulators,
// 16 chained K=4 WMMA steps per accumulator (K total = 64), full f32 precision.
// (Codegen confirmed in round 1: 64 v_wmma_f32_16x16x4_f32, clause-batched loads.)
// ---------------------------------------------------------------------------
__global__ void __launch_bounds__(256)
gemm_gelu_kernel(const float* __restrict__ x,
                 const float* __restrict__ W,
                 const float* __restrict__ b_lin,
                 float* __restrict__ u) {
    const int lane = threadIdx.x & 31;
    const int wave = (blockIdx.x * blockDim.x + threadIdx.x) >> 5;  // 0..16383
    const int b  = wave >> 10;        // 16 batches
    const int lt = wave & 1023;       // 1024 l-tiles of 16
    const int l0 = lt * 16;

    const int mn = lane & 15;         // M index for A frags, N index for B frags
    const int kh = lane >> 4;         // K half-select

    const float* __restrict__ xb = x + (size_t)b * Hdim * Ldim;

    v8f acc0 = {}, acc1 = {}, acc2 = {}, acc3 = {};

#pragma unroll
    for (int k0 = 0; k0 < Hdim; k0 += 4) {
        const int hrow = k0 + 2 * kh;
        v2f bf;
        bf.x = xb[(size_t)hrow * Ldim + l0 + mn];
        bf.y = xb[(size_t)(hrow + 1) * Ldim + l0 + mn];

        v2f a0, a1, a2, a3;
        a0.x = W[(0 * 16 + mn) * Hdim + hrow];
        a0.y = W[(0 * 16 + mn) * Hdim + hrow + 1];
        a1.x = W[(1 * 16 + mn) * Hdim + hrow];
        a1.y = W[(1 * 16 + mn) * Hdim + hrow + 1];
        a2.x = W[(2 * 16 + mn) * Hdim + hrow];
        a2.y = W[(2 * 16 + mn) * Hdim + hrow + 1];
        a3.x = W[(3 * 16 + mn) * Hdim + hrow];
        a3.y = W[(3 * 16 + mn) * Hdim + hrow + 1];

        acc0 = __builtin_amdgcn_wmma_f32_16x16x4_f32(false, a0, false, bf,
                                                     (short)0, acc0, false, false);
        acc1 = __builtin_amdgcn_wmma_f32_16x16x4_f32(false, a1, false, bf,
                                                     (short)0, acc1, false, false);
        acc2 = __builtin_amdgcn_wmma_f32_16x16x4_f32(false, a2, false, bf,
                                                     (short)0, acc2, false, false);
        acc3 = __builtin_amdgcn_wmma_f32_16x16x4_f32(false, a3, false, bf,
                                                     (short)0, acc3, false, false);
    }

    float* __restrict__ ub = u + (size_t)b * Hdim * Ldim;
#pragma unroll
    for (int r = 0; r < 8; ++r) {
        const int gr = r + kh * 8;
        const int l = l0 + mn;
        float v;
        v = acc0[r] + b_lin[0 * 16 + gr];
        ub[(size_t)(0 * 16 + gr) * Ldim + l] = gelu_exact(v);
        v = acc1[r] + b_lin[1 * 16 + gr];
        ub[(size_t)(1 * 16 + gr) * Ldim + l] = gelu_exact(v);
        v = acc2[r] + b_lin[2 * 16 + gr];
        ub[(size_t)(2 * 16 + gr) * Ldim + l] = gelu_exact(v);
        v = acc3[r] + b_lin[3 * 16 + gr];
        ub[(size_t)(3 * 16 + gr) * Ldim + l] = gelu_exact(v);
    }
}

// ---------------------------------------------------------------------------
// Kernel 4 (scan pass A): per-segment local end-states.
// Block = (bh, p); thread n scans u[bh, p*SEG .. p*SEG+SEG-1] from zero state,
// emitting only the final complex state E[bh,p,n]. Serial chain = SEG steps.
// ---------------------------------------------------------------------------
__global__ void __launch_bounds__(64)
s4d_state_kernel(const float* __restrict__ u,
                 const float* __restrict__ lam_re,
                 const float* __restrict__ lam_im,
                 float* __restrict__ E_re,
                 float* __restrict__ E_im) {
    const int blk = blockIdx.x;          // bh * Pdim + p
    const int bh = blk / Pdim;
    const int p  = blk % Pdim;
    const int h  = bh % Hdim;
    const int tid = threadIdx.x;         // mode n

    __shared__ float su[64];

    const float lr = lam_re[h * Ndim + tid];
    const float li = lam_im[h * Ndim + tid];

    const float* __restrict__ ub = u + (size_t)bh * Ldim + (size_t)p * SEG;

    float sre = 0.0f, sim = 0.0f;
    for (int c = 0; c < SEG / 64; ++c) {
        const int l0 = c * 64;
        su[tid] = ub[l0 + tid];
        if (l0 + 64 < SEG) __builtin_prefetch(ub + l0 + 64 + tid, 0, 0);
        __syncthreads();
#pragma unroll 16
        for (int i = 0; i < 64; ++i) {
            const float uv = su[i];
            const float nsre = fmaf(lr, sre, fmaf(-li, sim, uv));
            const float nsim = fmaf(lr, sim, li * sre);
            sre = nsre;
            sim = nsim;
        }
        __syncthreads();
    }
    const size_t e = ((size_t)bh * Pdim + p) * Ndim + tid;
    E_re[e] = sre;
    E_im[e] = sim;
}

// ---------------------------------------------------------------------------
// Kernel 5 (scan combine): prefix over the P=16 segments per (bh, n).
//   S_init[0] = 0;  S_init[p] = lamS * S_init[p-1] + E[p-1]
// ---------------------------------------------------------------------------
__global__ void s4d_combine_kernel(const float* __restrict__ E_re,
                                   const float* __restrict__ E_im,
                                   const float* __restrict__ lamS_re,
                                   const float* __restrict__ lamS_im,
                                   float* __restrict__ S_re,
                                   float* __restrict__ S_im) {
    int idx = blockIdx.x * blockDim.x + threadIdx.x;
    if (idx >= Bdim * Hdim * Ndim) return;
    const int bh = idx / Ndim;
    const int n  = idx % Ndim;
    const int h  = bh % Hdim;
    const float Lr = lamS_re[h * Ndim + n];
    const float Li = lamS_im[h * Ndim + n];
    float sr = 0.0f, si = 0.0f;
#pragma unroll
    for (int p = 0; p < Pdim; ++p) {
        const size_t o = ((size_t)bh * Pdim + p) * Ndim + n;
        S_re[o] = sr;
        S_im[o] = si;
        const float er = E_re[o], ei = E_im[o];
        const float nsr = fmaf(Lr, sr, fmaf(-Li, si, er));
        const float nsi = fmaf(Lr, si, fmaf(Li, sr, ei));
        sr = nsr;
        si = nsi;
    }
}

// ---------------------------------------------------------------------------
// Kernel 6 (scan pass B): rescan each segment from S_init, producing outputs
// with fused D-term, FiLM, GELU and residual. Chunks of 64 l's staged in LDS;
// per-mode contributions in sc[n][i] (padded rows), column-reduced per thread.
// ---------------------------------------------------------------------------
__global__ void __launch_bounds__(64)
s4d_scan_out_kernel(const float* __restrict__ u,
                    const float* __restrict__ x,
                    const float* __restrict__ lam_re,
                    const float* __restrict__ lam_im,
                    const float* __restrict__ c2_re,
                    const float* __restrict__ c2_im,
                    const float* __restrict__ S_re,
                    const float* __restrict__ S_im,
                    const float* __restrict__ Dv,
                    const float* __restrict__ g_film,
                    const float* __restrict__ b_film,
                    const float* __restrict__ res_w,
                    float* __restrict__ out) {
    const int blk = blockIdx.x;          // bh * Pdim + p
    const int bh = blk / Pdim;
    const int p  = blk % Pdim;
    const int b  = bh / Hdim;
    const int h  = bh % Hdim;
    const int tid = threadIdx.x;         // mode n

    __shared__ float su[64];
    __shared__ float sx[64];
    __shared__ float sc[64][65];         // padded: conflict-free column reads

    const int pn = h * Ndim + tid;
    const float lr = lam_re[pn];
    const float li = lam_im[pn];
    const float cr = c2_re[pn];
    const float ci = c2_im[pn];

    const float Dh = Dv[h];
    const float gg = g_film[b * Hdim + h];
    const float bb = b_film[b * Hdim + h];
    const float rw = res_w[h];

    const size_t base = (size_t)bh * Ldim + (size_t)p * SEG;
    const float* __restrict__ ub = u + base;
    const float* __restrict__ xb = x + base;
    float* __restrict__ ob = out + base;

    const size_t e = ((size_t)bh * Pdim + p) * Ndim + tid;
    float sre = S_re[e];
    float sim = S_im[e];

    for (int c = 0; c < SEG / 64; ++c) {
        const int l0 = c * 64;
        su[tid] = ub[l0 + tid];
        sx[tid] = xb[l0 + tid];
        if (l0 + 64 < SEG) {  // speculative prefetch of next chunk
            __builtin_prefetch(ub + l0 + 64 + tid, 0, 0);
            __builtin_prefetch(xb + l0 + 64 + tid, 0, 0);
        }
        __syncthreads();

        // 64 sequential recurrence steps for mode n = tid
#pragma unroll 8
        for (int i = 0; i < 64; ++i) {
            const float uv = su[i];
            const float nsre = fmaf(lr, sre, fmaf(-li, sim, uv));
            const float nsim = fmaf(lr, sim, li * sre);
            sre = nsre;
            sim = nsim;
            sc[tid][i] = fmaf(cr, sre, -ci * sim);  // 2*Re(coef * s)
        }
        __syncthreads();

        // thread i: reduce over the 64 modes, then fused epilogue
        float acc = Dh * su[tid];
#pragma unroll
        for (int n2 = 0; n2 < 64; ++n2) acc += sc[n2][tid];

        const float z = fmaf(acc, gg, bb);
        ob[l0 + tid] = gelu_exact(z) + sx[tid] * rw;
        __syncthreads();
    }
}

// ---------------------------------------------------------------------------
// Host-side launch
// ---------------------------------------------------------------------------
extern "C" void kernel_launch(void* const* d_in, const int* in_sizes, int n_in,
                              void* d_out, int out_size, void* d_ws, size_t ws_size,
                              hipStream_t stream) {
    const float* x          = (const float*)d_in[0];
    const float* cond       = (const float*)d_in[1];
    const float* W_lin      = (const float*)d_in[2];
    const float* b_lin      = (const float*)d_in[3];
    const float* log_dt     = (const float*)d_in[4];
    const float* log_A_real = (const float*)d_in[5];
    const float* A_imag     = (const float*)d_in[6];
    const float* C_re       = (const float*)d_in[7];
    const float* C_im       = (const float*)d_in[8];
    const float* Dv         = (const float*)d_in[9];
    const float* film_W     = (const float*)d_in[10];
    const float* film_b     = (const float*)d_in[11];
    const float* res_w      = (const float*)d_in[12];
    float* out = (float*)d_out;

    // Workspace layout
    char* ws = (char*)d_ws;
    const size_t uBytes  = (size_t)Bdim * Hdim * Ldim * sizeof(float);        // 64 MiB
    const size_t hnBytes = (size_t)Hdim * Ndim * sizeof(float);               // 16 KiB
    const size_t bhBytes = (size_t)Bdim * Hdim * sizeof(float);               // 4 KiB
    const size_t stBytes = (size_t)Bdim * Hdim * Pdim * Ndim * sizeof(float); // 4 MiB

    size_t off = 0;
    float* u       = (float*)(ws + off); off += uBytes;
    float* lam_re  = (float*)(ws + off); off += hnBytes;
    float* lam_im  = (float*)(ws + off); off += hnBytes;
    float* lamS_re = (float*)(ws + off); off += hnBytes;
    float* lamS_im = (float*)(ws + off); off += hnBytes;
    float* c2_re   = (float*)(ws + off); off += hnBytes;
    float* c2_im   = (float*)(ws + off); off += hnBytes;
    float* g_film  = (float*)(ws + off); off += bhBytes;
    float* b_film  = (float*)(ws + off); off += bhBytes;
    float* E_re    = (float*)(ws + off); off += stBytes;
    float* E_im    = (float*)(ws + off); off += stBytes;
    float* S_re    = (float*)(ws + off); off += stBytes;
    float* S_im    = (float*)(ws + off); off += stBytes;

    s4d_params_kernel<<<(Hdim * Ndim + 255) / 256, 256, 0, stream>>>(
        log_dt, log_A_real, A_imag, C_re, C_im,
        lam_re, lam_im, lamS_re, lamS_im, c2_re, c2_im);

    film_kernel<<<(Bdim * 2 * Hdim + 255) / 256, 256, 0, stream>>>(
        cond, film_W, film_b, g_film, b_film);

    // 16384 waves (one per (b, l-tile)), 8 waves/block
    gemm_gelu_kernel<<<(Bdim * (Ldim / 16)) / 8, 256, 0, stream>>>(
        x, W_lin, b_lin, u);

    // Pass A: local segment states (B*H*P blocks, 64 threads)
    s4d_state_kernel<<<Bdim * Hdim * Pdim, 64, 0, stream>>>(
        u, lam_re, lam_im, E_re, E_im);

    // Combine: 16-step prefix per (bh, n)
    s4d_combine_kernel<<<(Bdim * Hdim * Ndim + 255) / 256, 256, 0, stream>>>(
        E_re, E_im, lamS_re, lamS_im, S_re, S_im);

    // Pass B: outputs with fused epilogue
    s4d_scan_out_kernel<<<Bdim * Hdim * Pdim, 64, 0, stream>>>(
        u, x, lam_re, lam_im, c2_re, c2_im, S_re, S_im,
        Dv, g_film, b_film, res_w, out);
}